// DynamicEdgeConv_28595892257410
// MI455X (gfx1250) — compile-verified
//
#include <hip/hip_runtime.h>
#include <math.h>

// MI455X / gfx1250, wave32. fp32 WMMA (V_WMMA_F32_16X16X4_F32) for both GEMMs.

typedef float v2f __attribute__((ext_vector_type(2)));
typedef float v8f __attribute__((ext_vector_type(8)));

#define IN_CH 128
#define OUT_CH 128
#define HID 64
#define NPAD 80      // float2 slots per k-pair row: half-wave bank shift = 160%64 = 32 -> conflict-free b64
#define H_STRIDE 68  // per-wave 16x64 epilogue scratch, conflict-free writes

// ---------------------------------------------------------------------------
// xt = x @ conv_w   ([N,128] @ [128,128]); one wave per 16-node tile.
// CW staged in LDS as float2 K-pairs: slot(kp, n) -> one ds_load_b64 per B frag.
// ---------------------------------------------------------------------------
__global__ __launch_bounds__(256) void k_xt(const float* __restrict__ x,
                                            const float* __restrict__ cw,
                                            float* __restrict__ xt, int n_nodes) {
  __shared__ float CWf[64 * NPAD * 2];           // 40 KB
  int tid = threadIdx.x;
  for (int i = tid; i < IN_CH * OUT_CH; i += 256) {
    int k = i >> 7, n = i & 127;
    CWf[((k >> 1) * NPAD + n) * 2 + (k & 1)] = cw[i];
  }
  __syncthreads();
  const v2f* CWp = (const v2f*)CWf;

  int lane = tid & 31, h = lane >> 4, nl = lane & 15;
  int wave = blockIdx.x * 8 + (tid >> 5);
  int nwaves = gridDim.x * 8;
  int ntiles = (n_nodes + 15) >> 4;

  for (int t = wave; t < ntiles; t += nwaves) {
    int mrow = t * 16 + nl;
    if (mrow > n_nodes - 1) mrow = n_nodes - 1;   // clamp keeps EXEC all-ones
    const float* xrow = x + (size_t)mrow * IN_CH;
    v8f acc[8] = {};
    v2f a = *(const v2f*)(xrow + 2 * h);          // A fragment for ks=0
#pragma unroll 2
    for (int ks = 0; ks < 32; ++ks) {
      int ksn = (ks < 31) ? ks + 1 : 31;          // branchless A prefetch (no data phi)
      v2f na = *(const v2f*)(xrow + 4 * ksn + 2 * h);
      const v2f* bp = CWp + (2 * ks + h) * NPAD + nl;
      v2f bfrag[8];
#pragma unroll
      for (int nt = 0; nt < 8; ++nt) bfrag[nt] = bp[nt * 16];   // batch DS issues
#pragma unroll
      for (int nt = 0; nt < 8; ++nt)
        acc[nt] = __builtin_amdgcn_wmma_f32_16x16x4_f32(
            false, a, false, bfrag[nt], (short)0, acc[nt], false, false);
      a = na;
    }
    if (t * 16 + 16 <= n_nodes) {                 // uniform fast path: no guards
#pragma unroll
      for (int nt = 0; nt < 8; ++nt)
#pragma unroll
        for (int v = 0; v < 8; ++v)
          xt[(size_t)(t * 16 + v + 8 * h) * OUT_CH + nt * 16 + nl] = acc[nt][v];
    } else {
#pragma unroll
      for (int nt = 0; nt < 8; ++nt)
#pragma unroll
        for (int v = 0; v < 8; ++v) {
          int m = t * 16 + v + 8 * h;
          if (m < n_nodes) xt[(size_t)m * OUT_CH + nt * 16 + nl] = acc[nt][v];
        }
    }
  }
}

// ---------------------------------------------------------------------------
// Edge MLP: ew[e] = sigmoid( GELU(concat(x[row],x[col]) @ W1 + b1) @ W2 + b2 )
// One wave per PAIR of 16-edge tiles (M=32): B fragments shared by both tiles.
// W1 (256x64) staged in LDS as float2 K-pairs (128 x NPAD slots).
// ---------------------------------------------------------------------------
__global__ __launch_bounds__(256) void k_edge_mlp(
    const float* __restrict__ x, const int* __restrict__ ei,
    const float* __restrict__ w1, const float* __restrict__ b1,
    const float* __restrict__ w2, const float* __restrict__ b2,
    float* __restrict__ ew, int n_edges) {
  extern __shared__ float smem[];
  float* W1f = smem;                          // 128 * NPAD * 2 floats (80 KB)
  float* W2s = W1f + 128 * NPAD * 2;          // 64
  float* B1s = W2s + HID;                     // 64
  float* Hs  = B1s + HID;                     // 8 waves * 16 * H_STRIDE

  int tid = threadIdx.x;
  for (int i = tid; i < 256 * HID; i += 256) {
    int k = i >> 6, n = i & 63;
    W1f[((k >> 1) * NPAD + n) * 2 + (k & 1)] = w1[i];
  }
  if (tid < HID) { W2s[tid] = w2[tid]; B1s[tid] = b1[tid]; }
  __syncthreads();
  const v2f* W1p = (const v2f*)W1f;

  float b2v = b2[0];
  int lane = tid & 31, h = lane >> 4, nl = lane & 15;
  int wv = tid >> 5;
  float* Hw = Hs + wv * 16 * H_STRIDE;        // private per-wave scratch
  int wave = blockIdx.x * 8 + wv;
  int nwaves = gridDim.x * 8;
  int ntiles = (n_edges + 15) >> 4;
  int npairs = (ntiles + 1) >> 1;

  for (int p = wave; p < npairs; p += nwaves) {
    int t0 = 2 * p, t1 = 2 * p + 1;
    int e0 = t0 * 16 + nl, e1 = t1 * 16 + nl;
    int ec0 = (e0 > n_edges - 1) ? (n_edges - 1) : e0;
    int ec1 = (e1 > n_edges - 1) ? (n_edges - 1) : e1;
    const float* xr0 = x + (size_t)ei[ec0] * IN_CH;
    const float* xc0 = x + (size_t)ei[n_edges + ec0] * IN_CH;
    const float* xr1 = x + (size_t)ei[ec1] * IN_CH;
    const float* xc1 = x + (size_t)ei[n_edges + ec1] * IN_CH;

    v8f acc0[4] = {}, acc1[4] = {};
#pragma unroll 1
    for (int half = 0; half < 2; ++half) {      // concat halves: x[row], x[col]
      const float* c0 = half ? xc0 : xr0;
      const float* c1 = half ? xc1 : xr1;
      int kpb = half * 64;                      // k-pair base
      v2f a0 = *(const v2f*)(c0 + 2 * h);
      v2f a1 = *(const v2f*)(c1 + 2 * h);
#pragma unroll 2
      for (int ks = 0; ks < 32; ++ks) {
        int ksn = (ks < 31) ? ks + 1 : 31;      // branchless A prefetch (no data phi)
        v2f na0 = *(const v2f*)(c0 + 4 * ksn + 2 * h);
        v2f na1 = *(const v2f*)(c1 + 4 * ksn + 2 * h);
        const v2f* bp = W1p + (kpb + 2 * ks + h) * NPAD + nl;
        v2f bfrag[4];
#pragma unroll
        for (int nt = 0; nt < 4; ++nt) bfrag[nt] = bp[nt * 16];  // batch DS issues
#pragma unroll
        for (int nt = 0; nt < 4; ++nt) {
          acc0[nt] = __builtin_amdgcn_wmma_f32_16x16x4_f32(
              false, a0, false, bfrag[nt], (short)0, acc0[nt], false, false);
          acc1[nt] = __builtin_amdgcn_wmma_f32_16x16x4_f32(
              false, a1, false, bfrag[nt], (short)0, acc1[nt], false, false);
        }
        a0 = na0; a1 = na1;
      }
    }
    // Epilogue per tile: bias + exact GELU -> LDS scratch -> W2 reduction.
#pragma unroll 1
    for (int sub = 0; sub < 2; ++sub) {
      v8f* accp = sub ? acc1 : acc0;
      int e = sub ? e1 : e0;
#pragma unroll
      for (int nt = 0; nt < 4; ++nt) {
        float bn = B1s[nt * 16 + nl];
#pragma unroll
        for (int v = 0; v < 8; ++v) {
          float hv = accp[nt][v] + bn;
          hv = 0.5f * hv * (1.0f + erff(hv * 0.70710678118654752f));
          Hw[(v + 8 * h) * H_STRIDE + nt * 16 + nl] = hv;   // row = v + 8h
        }
      }
      float s = 0.0f;                            // LDS ops in-order within a wave
#pragma unroll
      for (int j = 0; j < 32; ++j)
        s += Hw[nl * H_STRIDE + 32 * h + j] * W2s[32 * h + j];
      s += __shfl_xor(s, 16, 32);
      if (lane < 16 && e < n_edges) {
        float z = s + b2v;
        ew[e] = 1.0f / (1.0f + expf(-z));
      }
    }
  }
}

// ---------------------------------------------------------------------------
// Normalization + aggregation helpers
// ---------------------------------------------------------------------------
__global__ void k_deg_init(float* __restrict__ deg, int n) {
  int i = blockIdx.x * blockDim.x + threadIdx.x;
  if (i < n) deg[i] = 1.0f;                      // self-loop weight
}

__global__ void k_deg_acc(const int* __restrict__ ei, const float* __restrict__ ew,
                          float* __restrict__ deg, int n_edges) {
  int e = blockIdx.x * blockDim.x + threadIdx.x;
  if (e < n_edges) atomicAdd(&deg[ei[n_edges + e]], ew[e]);
}

__global__ void k_dinv(const float* __restrict__ deg, float* __restrict__ dinv, int n) {
  int i = blockIdx.x * blockDim.x + threadIdx.x;
  if (i < n) { float d = deg[i]; dinv[i] = d > 0.0f ? rsqrtf(d) : 0.0f; }
}

// out = conv_b + dinv^2 * xt   (self-loop term; fully initializes d_out)
__global__ void k_self(const float* __restrict__ xt, const float* __restrict__ dinv,
                       const float* __restrict__ cb, float* __restrict__ out, int n) {
  size_t i = (size_t)blockIdx.x * blockDim.x + threadIdx.x;
  if (i < (size_t)n * OUT_CH) {
    int node = (int)(i >> 7), ch = (int)(i & 127);
    float d = dinv[node];
    out[i] = cb[ch] + d * d * xt[i];
  }
}

// ew[e] <- dinv[row]*ew[e]*dinv[col]
__global__ void k_norm(const int* __restrict__ ei, const float* __restrict__ dinv,
                       float* __restrict__ ew, int n_edges) {
  int e = blockIdx.x * blockDim.x + threadIdx.x;
  if (e < n_edges) ew[e] = ew[e] * dinv[ei[e]] * dinv[ei[n_edges + e]];
}

// out[col] += norm * xt[row] ; one wave per edge, 4 channels per lane
__global__ void k_scatter(const int* __restrict__ ei, const float* __restrict__ ew,
                          const float* __restrict__ xt, float* __restrict__ out,
                          int n_edges) {
  size_t tid = (size_t)blockIdx.x * blockDim.x + threadIdx.x;
  int e = (int)(tid >> 5);
  if (e >= n_edges) return;
  int lane = (int)(tid & 31);
  int r = ei[e], c = ei[n_edges + e];
  float w = ew[e];
  const float4 v = *(const float4*)(xt + (size_t)r * OUT_CH + lane * 4);
  float* dst = out + (size_t)c * OUT_CH + lane * 4;
  atomicAdd(dst + 0, w * v.x);
  atomicAdd(dst + 1, w * v.y);
  atomicAdd(dst + 2, w * v.z);
  atomicAdd(dst + 3, w * v.w);
}

// ---------------------------------------------------------------------------
extern "C" void kernel_launch(void* const* d_in, const int* in_sizes, int n_in,
                              void* d_out, int out_size, void* d_ws, size_t ws_size,
                              hipStream_t stream) {
  const float* x  = (const float*)d_in[0];
  const int*   ei = (const int*)d_in[1];     // int32 per harness contract
  const float* w1 = (const float*)d_in[2];
  const float* b1 = (const float*)d_in[3];
  const float* w2 = (const float*)d_in[4];
  const float* b2 = (const float*)d_in[5];
  const float* cw = (const float*)d_in[6];
  const float* cb = (const float*)d_in[7];
  int n_nodes = in_sizes[0] / IN_CH;   // 50000
  int n_edges = in_sizes[1] / 2;       // 800000
  float* out = (float*)d_out;

  float* xt   = (float*)d_ws;                       // N*128
  float* ew   = xt + (size_t)n_nodes * OUT_CH;      // E
  float* deg  = ew + n_edges;                       // N
  float* dinv = deg + n_nodes;                      // N

  int node_tiles = (n_nodes + 15) / 16;
  int edge_tiles = (n_edges + 15) / 16;
  int edge_pairs = (edge_tiles + 1) / 2;
  int g_xt  = (node_tiles + 7) / 8;                 // 1 tile per wave
  int g_mlp = ((edge_pairs + 7) / 8 + 7) / 8;       // ~8 pairs per wave
  if (g_mlp < 1) g_mlp = 1;
  size_t mlp_lds = (size_t)(128 * NPAD * 2 + 2 * HID + 8 * 16 * H_STRIDE) * sizeof(float);

  k_xt<<<g_xt, 256, 0, stream>>>(x, cw, xt, n_nodes);
  k_edge_mlp<<<g_mlp, 256, mlp_lds, stream>>>(x, ei, w1, b1, w2, b2, ew, n_edges);

  int bN = (n_nodes + 255) / 256;
  int bE = (n_edges + 255) / 256;
  k_deg_init<<<bN, 256, 0, stream>>>(deg, n_nodes);
  k_deg_acc<<<bE, 256, 0, stream>>>(ei, ew, deg, n_edges);
  k_dinv<<<bN, 256, 0, stream>>>(deg, dinv, n_nodes);
  size_t total = (size_t)n_nodes * OUT_CH;
  k_self<<<(int)((total + 255) / 256), 256, 0, stream>>>(xt, dinv, cb, out, n_nodes);
  k_norm<<<bE, 256, 0, stream>>>(ei, dinv, ew, n_edges);
  size_t sth = (size_t)n_edges * 32;
  k_scatter<<<(int)((sth + 255) / 256), 256, 0, stream>>>(ei, ew, xt, out, n_edges);
}